// PathPredictor_86560770884084
// MI455X (gfx1250) — compile-verified
//
#include <hip/hip_runtime.h>
#include <math.h>

// ---------------- problem constants ----------------
#define DD   200          // d
#define RR   500          // num_rel
#define NN_  20000        // num entities
#define BB   2048         // batch
#define TT   32           // timesteps
#define NEGV (-1.0e9f)

// GEMM tiling
#define MB   64           // block M tile
#define NB   64           // block N tile
#define KC   32           // K chunk staged in LDS
#define LDT  34           // padded LDS K-stride (even => 8B-aligned b64, conflict-free)

typedef __attribute__((ext_vector_type(2))) float v2f;
typedef __attribute__((ext_vector_type(8))) float v8f;

__device__ __forceinline__ float sigmoidf_(float x) { return 1.0f / (1.0f + expf(-x)); }

// =====================================================================
// LDS double-buffered WMMA f32 GEMM:
//   C[M,N] = A[M,K] @ (TRANSB ? W[N,K]^T : B[K,N]) + bias
// Block = 128 threads (4 waves) computes a 64x64 C tile; wave w owns rows
// [w*16, w*16+16) x 64 cols (4 accumulators). K is consumed in 32-wide chunks
// staged in LDS: As row-major (m*LDT+k), Bs col-major (n*LDT+k) so each
// fragment pair (k, k+1) is one ds_load_b64. Chunk c+1 is fetched (coalesced,
// register-staged) while chunk c is multiplied -> global latency overlaps WMMA.
// =====================================================================
template <bool TRANSB>
__global__ __launch_bounds__(128)
void gemm_wmma_lds(const float* __restrict__ A, const float* __restrict__ Bm,
                   const float* __restrict__ bias, float* __restrict__ C,
                   int M, int N, int K) {
  __shared__ float As[2][MB * LDT];
  __shared__ float Bs[2][NB * LDT];

  const int t    = threadIdx.x;
  const int lane = t & 31, wave = t >> 5;
  const int hf   = lane >> 4, lm = lane & 15;
  const int nbn  = (N + NB - 1) / NB;
  const int m0   = (blockIdx.x / nbn) * MB;
  const int n0   = (blockIdx.x % nbn) * NB;

  const int nchunks = (K + KC - 1) / KC;

  // cooperative, coalesced copy of one K-chunk into LDS buffer `buf`
  auto copy_chunk = [&](int buf, int k0) {
    float ta[16], tb[16];
#pragma unroll
    for (int i = 0; i < 16; ++i) {            // A: 64x32, lane-consecutive cols
      const int flat = t + i * 128;
      const int r = flat >> 5, c = flat & 31;
      const bool ok = (m0 + r) < M && (k0 + c) < K;
      ta[i] = ok ? A[(size_t)(m0 + r) * K + k0 + c] : 0.0f;
    }
#pragma unroll
    for (int i = 0; i < 16; ++i) {            // B: 32x64
      const int flat = t + i * 128;
      if (TRANSB) {                           // read W rows (consecutive k)
        const int n = flat >> 5, c = flat & 31;
        const bool ok = (n0 + n) < N && (k0 + c) < K;
        tb[i] = ok ? Bm[(size_t)(n0 + n) * K + k0 + c] : 0.0f;
      } else {                                // read B rows (consecutive n)
        const int c = flat >> 6, n = flat & 63;
        const bool ok = (k0 + c) < K && (n0 + n) < N;
        tb[i] = ok ? Bm[(size_t)(k0 + c) * N + n0 + n] : 0.0f;
      }
    }
#pragma unroll
    for (int i = 0; i < 16; ++i) {
      const int flat = t + i * 128;
      const int r = flat >> 5, c = flat & 31;
      As[buf][r * LDT + c] = ta[i];
    }
#pragma unroll
    for (int i = 0; i < 16; ++i) {
      const int flat = t + i * 128;
      if (TRANSB) { const int n = flat >> 5, c = flat & 31; Bs[buf][n * LDT + c] = tb[i]; }
      else        { const int c = flat >> 6, n = flat & 63; Bs[buf][n * LDT + c] = tb[i]; }
    }
  };

  copy_chunk(0, 0);
  v8f acc[4] = {};
  const int arow = wave * 16 + lm;

  for (int c = 0; c < nchunks; ++c) {
    __syncthreads();                          // buffer c&1 ready; c-1 compute done
    if (c + 1 < nchunks) copy_chunk((c + 1) & 1, (c + 1) * KC);
    const int buf = c & 1;
#pragma unroll
    for (int kk = 0; kk < KC; kk += 4) {
      const int ka = kk + hf * 2;
      const v2f a = *(const v2f*)&As[buf][arow * LDT + ka];
#pragma unroll
      for (int j = 0; j < 4; ++j) {
        const v2f b = *(const v2f*)&Bs[buf][(j * 16 + lm) * LDT + ka];
        acc[j] = __builtin_amdgcn_wmma_f32_16x16x4_f32(false, a, false, b,
                                                       (short)0, acc[j], false, false);
      }
    }
  }

#pragma unroll
  for (int j = 0; j < 4; ++j) {
    const int col = n0 + j * 16 + lm;
    if (col >= N) continue;
    const float bv = bias ? bias[col] : 0.0f;
#pragma unroll
    for (int i = 0; i < 8; ++i) {
      const int r = m0 + wave * 16 + (hf ? i + 8 : i);
      if (r < M) C[(size_t)r * N + col] = acc[j][i] + bv;
    }
  }
}

// =====================================================================
// Fused bilinear: out[b,e] = sum_{kq,dd} u[b,kq] * x[b,dd] * Wtk[kq,dd,e]
// == GEMM (M=BB, K=DD*DD, N=DD) with A = outer(u_b, x_b) built in registers
// (avoids the reference's 328MB W_mat). The 4 waves of a block share one
// double-buffered LDS tile of Wtk (32 dd-rows x 208 cols, col-major) -> 4x
// less global Wtk traffic, fragments via ds_load_b64, Wtk (32MB) L2-resident.
// =====================================================================
__global__ __launch_bounds__(128)
void bilinear_wmma(const float* __restrict__ u, const float* __restrict__ x,
                   const float* __restrict__ Wtk, float* __restrict__ out) {
  __shared__ float Bs[2][208 * LDT];          // Bs[col*LDT + kk]
  const int t    = threadIdx.x;
  const int lane = t & 31, wave = t >> 5;
  const int hf   = lane >> 4, lm = lane & 15;
  const int strip = blockIdx.x * 4 + wave;    // 128 strips of 16 rows
  const int row   = strip * 16 + lm;

  // zero the pad columns (200..207) once; copies never touch them
  for (int i = t; i < 8 * KC; i += 128) {
    const int col = 200 + i / KC, k = i % KC;
    Bs[0][col * LDT + k] = 0.0f;
    Bs[1][col * LDT + k] = 0.0f;
  }

  const int NCH = DD * 7;                     // 7 dd-chunks of 32 per kq (last padded)
  auto copy_chunk = [&](int buf, int ch) {
    const int kq = ch / 7, dd0 = (ch % 7) * KC;
    const float* Wk = Wtk + (size_t)kq * DD * DD;
    if (ch + 1 < NCH)                          // gfx1250 global_prefetch_b8
      __builtin_prefetch(Wtk + (size_t)((ch + 1) / 7) * DD * DD + ((ch + 1) % 7) * KC * DD + t * 16, 0, 1);
#pragma unroll 5
    for (int g = 0; g < 5; ++g) {             // 6400 elems in 5 groups of 10
      float tv[10];
#pragma unroll
      for (int i = 0; i < 10; ++i) {
        const int flat = t + (g * 10 + i) * 128;
        const int col = flat % DD, kk = flat / DD;
        const int dd = dd0 + kk;
        tv[i] = (dd < DD) ? Wk[(size_t)dd * DD + col] : 0.0f;
      }
#pragma unroll
      for (int i = 0; i < 10; ++i) {
        const int flat = t + (g * 10 + i) * 128;
        const int col = flat % DD, kk = flat / DD;
        Bs[buf][col * LDT + kk] = tv[i];
      }
    }
  };

  copy_chunk(0, 0);
  v8f acc[13] = {};

  for (int c = 0; c < NCH; ++c) {
    __syncthreads();
    if (c + 1 < NCH) copy_chunk((c + 1) & 1, c + 1);
    const int buf = c & 1;
    const int kq = c / 7, dd0 = (c % 7) * KC;
    const float uv = u[(size_t)row * DD + kq];
#pragma unroll
    for (int kk = 0; kk < KC; kk += 4) {
      const int ka = kk + hf * 2;
      const int dd = dd0 + ka;
      v2f a;                                  // A[m][k] = u[m][kq] * x[m][dd]
      a.x = (dd     < DD) ? uv * x[(size_t)row * DD + dd]     : 0.0f;
      a.y = (dd + 1 < DD) ? uv * x[(size_t)row * DD + dd + 1] : 0.0f;
#pragma unroll
      for (int j = 0; j < 13; ++j) {
        const v2f b = *(const v2f*)&Bs[buf][(j * 16 + lm) * LDT + ka];
        acc[j] = __builtin_amdgcn_wmma_f32_16x16x4_f32(false, a, false, b,
                                                       (short)0, acc[j], false, false);
      }
    }
  }

#pragma unroll
  for (int j = 0; j < 13; ++j) {
    const int col = j * 16 + lm;
    if (col >= DD) continue;
#pragma unroll
    for (int i = 0; i < 8; ++i) {
      const int r = strip * 16 + (hf ? i + 8 : i);
      out[(size_t)r * DD + col] = acc[j][i];
    }
  }
}

// ---------------- elementwise / reduction kernels ----------------

__global__ void fill_zero(float* p, int n) {
  int i = blockIdx.x * blockDim.x + threadIdx.x;
  if (i < n) p[i] = 0.0f;
}

__global__ void gather_rows(const float* __restrict__ src, const int* __restrict__ idx,
                            int col_sel, float* __restrict__ dst, int rows, int cols) {
  int i = blockIdx.x * blockDim.x + threadIdx.x;
  if (i >= rows * cols) return;
  int r = i / cols, c = i % cols;
  dst[i] = src[(size_t)idx[r * 3 + col_sel] * cols + c];
}

// masked softmax over rel dim: attn[b,r] = softmax_r( (cur*qk)==0 ? NEG : cur*qk )
__global__ void softmax_attn(const float* __restrict__ pemb, const float* __restrict__ qk,
                             float* __restrict__ attn, int t) {
  const int b = blockIdx.x;
  const float* cur = pemb + (size_t)b * (TT * RR) + (size_t)t * RR;
  const float* qkb = qk + (size_t)b * RR;
  __shared__ float red[256];
  float mx = -INFINITY;
  for (int r = threadIdx.x; r < RR; r += 256) {
    float l = cur[r] * qkb[r];
    l = (l == 0.0f) ? NEGV : l;
    mx = fmaxf(mx, l);
  }
  red[threadIdx.x] = mx; __syncthreads();
  for (int s = 128; s > 0; s >>= 1) {
    if (threadIdx.x < s) red[threadIdx.x] = fmaxf(red[threadIdx.x], red[threadIdx.x + s]);
    __syncthreads();
  }
  mx = red[0]; __syncthreads();
  float sum = 0.0f;
  for (int r = threadIdx.x; r < RR; r += 256) {
    float l = cur[r] * qkb[r];
    l = (l == 0.0f) ? NEGV : l;
    sum += expf(l - mx);
  }
  red[threadIdx.x] = sum; __syncthreads();
  for (int s = 128; s > 0; s >>= 1) {
    if (threadIdx.x < s) red[threadIdx.x] += red[threadIdx.x + s];
    __syncthreads();
  }
  const float inv = 1.0f / red[0];
  for (int r = threadIdx.x; r < RR; r += 256) {
    float l = cur[r] * qkb[r];
    l = (l == 0.0f) ? NEGV : l;
    attn[(size_t)b * RR + r] = expf(l - mx) * inv;
  }
}

// GRU gate combine: h' = (1-z)*n + z*h
__global__ void gru_combine(const float* __restrict__ gi, const float* __restrict__ gh,
                            const float* __restrict__ hin, float* __restrict__ hout) {
  int i = blockIdx.x * blockDim.x + threadIdx.x;
  if (i >= BB * DD) return;
  int b = i / DD, j = i % DD;
  const float* gib = gi + (size_t)b * 3 * DD;
  const float* ghb = gh + (size_t)b * 3 * DD;
  float r = sigmoidf_(gib[j] + ghb[j]);
  float z = sigmoidf_(gib[DD + j] + ghb[DD + j]);
  float n = tanhf(gib[2 * DD + j] + r * ghb[2 * DD + j]);
  hout[i] = (1.0f - z) * n + z * hin[i];
}

__global__ void axpby(const float* __restrict__ a, const float* __restrict__ b,
                      float* __restrict__ c, float alpha, int n) {
  int i = blockIdx.x * blockDim.x + threadIdx.x;
  if (i < n) c[i] = alpha * a[i] + b[i];
}

__global__ void sqdiff_partial(const float* __restrict__ a, const float* __restrict__ b,
                               int n, float* __restrict__ part) {
  __shared__ float red[256];
  float s = 0.0f;
  for (int i = blockIdx.x * blockDim.x + threadIdx.x; i < n; i += gridDim.x * blockDim.x) {
    float d = a[i] - b[i];
    s += d * d;
  }
  red[threadIdx.x] = s; __syncthreads();
  for (int st = 128; st > 0; st >>= 1) {
    if (threadIdx.x < st) red[threadIdx.x] += red[threadIdx.x + st];
    __syncthreads();
  }
  if (threadIdx.x == 0) part[blockIdx.x] = red[0];
}

// BCE partial sums + in-place sigmoid of the score matrix
__global__ void bce_sigmoid(float* __restrict__ score, const int* __restrict__ triples,
                            float* __restrict__ part) {
  __shared__ float red[256];
  const int n = BB * NN_;
  float s = 0.0f;
  for (int i = blockIdx.x * blockDim.x + threadIdx.x; i < n; i += gridDim.x * blockDim.x) {
    int b = i / NN_, j = i % NN_;
    float v = score[i];
    // softplus(v) = max(v,0) + log1p(exp(-|v|)); softplus(-v) = softplus(v) - v
    float sp = fmaxf(v, 0.0f) + log1pf(expf(-fabsf(v)));
    s += (triples[b * 3 + 2] == j) ? (sp - v) : sp;
    score[i] = 1.0f / (1.0f + expf(-v));
  }
  red[threadIdx.x] = s; __syncthreads();
  for (int st = 128; st > 0; st >>= 1) {
    if (threadIdx.x < st) red[threadIdx.x] += red[threadIdx.x + st];
    __syncthreads();
  }
  if (threadIdx.x == 0) part[blockIdx.x] = red[0];
}

__global__ void reduce_final(const float* __restrict__ part, int n, float scale,
                             float* __restrict__ out) {
  __shared__ float red[256];
  float s = 0.0f;
  for (int i = threadIdx.x; i < n; i += 256) s += part[i];
  red[threadIdx.x] = s; __syncthreads();
  for (int st = 128; st > 0; st >>= 1) {
    if (threadIdx.x < st) red[threadIdx.x] += red[threadIdx.x + st];
    __syncthreads();
  }
  if (threadIdx.x == 0) *out = red[0] * scale;
}

// =====================================================================
extern "C" void kernel_launch(void* const* d_in, const int* in_sizes, int n_in,
                              void* d_out_v, int out_size, void* d_ws, size_t ws_size,
                              hipStream_t stream) {
  (void)in_sizes; (void)n_in; (void)out_size; (void)ws_size;
  const float* pre_emb = (const float*)d_in[0];
  const float* r_emb   = (const float*)d_in[1];
  const float* pemb    = (const float*)d_in[2];
  const int*   triples = (const int*)d_in[3];
  // d_in[4]=num_rel, d_in[5]=cur_ts (compile-time constants here)
  const float* w_map1 = (const float*)d_in[6];
  const float* b_map1 = (const float*)d_in[7];
  const float* w_map2 = (const float*)d_in[8];
  const float* b_map2 = (const float*)d_in[9];
  const float* w_r2h  = (const float*)d_in[10];
  const float* b_r2h  = (const float*)d_in[11];
  const float* w_r2h2 = (const float*)d_in[12];
  const float* b_r2h2 = (const float*)d_in[13];
  const float* w_align= (const float*)d_in[14];
  const float* b_align= (const float*)d_in[15];
  const float* w_gruh = (const float*)d_in[16];
  const float* b_gruh = (const float*)d_in[17];
  const float* w_attn = (const float*)d_in[18];
  const float* b_attn = (const float*)d_in[19];
  const float* w_ih   = (const float*)d_in[20];
  const float* w_hh   = (const float*)d_in[21];
  const float* b_ih   = (const float*)d_in[22];
  const float* b_hh   = (const float*)d_in[23];
  const float* Wtk    = (const float*)d_in[24];

  float* out   = (float*)d_out_v;     // [0]=match_loss, [1]=path_loss, [2..]=score
  float* score = out + 2;

  // ---- workspace carve (~55 MB of f32) ----
  float* p = (float*)d_ws;
  auto take = [&](size_t n) { float* r = p; p += n; return r; };
  float* hidden1 = take((size_t)RR * 2 * DD);   // 500x400
  float* mrel    = take((size_t)RR * DD);       // 500x200
  float* qrel    = take((size_t)BB * DD);
  float* attnq   = take((size_t)BB * DD);
  float* qk      = take((size_t)BB * RR);
  float* attn    = take((size_t)BB * RR);
  float* h       = take((size_t)BB * DD);
  float* relpath = take((size_t)BB * DD);
  float* gi      = take((size_t)BB * 3 * DD);
  float* ghg     = take((size_t)BB * 3 * DD);
  float* t1      = take((size_t)BB * DD);
  float* t2      = take((size_t)BB * DD);
  float* ph      = take((size_t)BB * DD);
  float* ghb     = take((size_t)BB * DD);
  float* sub     = take((size_t)BB * DD);
  float* x1      = take((size_t)BB * DD);
  float* uu      = take((size_t)BB * DD);
  float* x2      = take((size_t)BB * DD);
  float* palign  = take((size_t)NN_ * DD);
  float* partA   = take(256);
  float* partB   = take(4096);

  auto gemm_nt = [&](const float* A, const float* W, const float* bias, float* C,
                     int M, int N, int K) {
    int blocks = ((M + MB - 1) / MB) * ((N + NB - 1) / NB);
    gemm_wmma_lds<true><<<blocks, 128, 0, stream>>>(A, W, bias, C, M, N, K);
  };
  auto gemm_nn = [&](const float* A, const float* B, const float* bias, float* C,
                     int M, int N, int K) {
    int blocks = ((M + MB - 1) / MB) * ((N + NB - 1) / NB);
    gemm_wmma_lds<false><<<blocks, 128, 0, stream>>>(A, B, bias, C, M, N, K);
  };
  const int EW = (BB * DD + 255) / 256;   // elementwise grid over B*d

  // ---- mapped_rel = Linear2(Linear1(r_emb)) ----
  gemm_nt(r_emb, w_map1, b_map1, hidden1, RR, 2 * DD, DD);
  gemm_nt(hidden1, w_map2, b_map2, mrel, RR, DD, 2 * DD);

  // ---- q_rel gather + qk = Linear_attn(q_rel) @ mapped_rel^T ----
  gather_rows<<<EW, 256, 0, stream>>>(mrel, triples, 1, qrel, BB, DD);
  gemm_nt(qrel, w_attn, b_attn, attnq, BB, DD, DD);
  gemm_nt(attnq, mrel, nullptr, qk, BB, RR, DD);

  // ---- GRU scan over T timesteps ----
  fill_zero<<<EW, 256, 0, stream>>>(h, BB * DD);
  for (int t = 0; t < TT; ++t) {
    softmax_attn<<<BB, 256, 0, stream>>>(pemb, qk, attn, t);
    gemm_nn(attn, mrel, nullptr, relpath, BB, DD, RR);
    gemm_nt(relpath, w_ih, b_ih, gi, BB, 3 * DD, DD);
    gemm_nt(h, w_hh, b_hh, ghg, BB, 3 * DD, DD);
    gru_combine<<<EW, 256, 0, stream>>>(gi, ghg, h, h);
  }

  // ---- predicted_hist = 0.1*Linear2(Linear1(q_rel)) + q_rel; match_loss ----
  gemm_nt(qrel, w_r2h, b_r2h, t1, BB, DD, DD);
  gemm_nt(t1, w_r2h2, b_r2h2, t2, BB, DD, DD);
  axpby<<<EW, 256, 0, stream>>>(t2, qrel, ph, 0.1f, BB * DD);
  sqdiff_partial<<<256, 256, 0, stream>>>(ph, h, BB * DD, partA);
  reduce_final<<<1, 256, 0, stream>>>(partA, 256, 1.0f / (float)(BB * DD), out + 0);

  // ---- gh = GRUCell(q_rel, predicted_hist) ----
  gemm_nt(qrel, w_ih, b_ih, gi, BB, 3 * DD, DD);
  gemm_nt(ph, w_hh, b_hh, ghg, BB, 3 * DD, DD);
  gru_combine<<<EW, 256, 0, stream>>>(gi, ghg, ph, ghb);

  // ---- bilinear score head ----
  gemm_nt(ghb, w_gruh, b_gruh, uu, BB, DD, DD);
  gather_rows<<<EW, 256, 0, stream>>>(pre_emb, triples, 0, sub, BB, DD);
  gemm_nt(sub, w_align, b_align, x1, BB, DD, DD);
  bilinear_wmma<<<BB / 16 / 4, 128, 0, stream>>>(uu, x1, Wtk, x2);
  gemm_nt(pre_emb, w_align, b_align, palign, NN_, DD, DD);
  gemm_nt(x2, palign, nullptr, score, BB, NN_, DD);

  // ---- BCE path_loss + sigmoid(score) in place ----
  bce_sigmoid<<<4096, 256, 0, stream>>>(score, triples, partB);
  reduce_final<<<1, 256, 0, stream>>>(partB, 4096, 1.0f / ((float)BB * (float)NN_), out + 1);
}